// BPTTDiagonalRNN_64510408786263
// MI455X (gfx1250) — compile-verified
//
#include <hip/hip_runtime.h>
#include <hip/hip_bf16.h>

typedef __attribute__((ext_vector_type(2))) float v2f;
typedef __attribute__((ext_vector_type(4))) float v4f;
typedef __attribute__((ext_vector_type(8))) float v8f;
typedef int gv4i __attribute__((vector_size(16)));   // matches builtin param pointee

#define TM 128          // output tile rows per workgroup
#define TN 128          // output tile cols per workgroup
#define KB 32           // K slice staged in LDS
#define LDSROW 36       // padded LDS row stride (dwords): 144B rows -> 16B aligned, conflict-free
#define CHUNK 128       // scan chunk length along T

#if __has_builtin(__builtin_amdgcn_global_load_async_to_lds_b128)
#define ASYNC_LDS 1
#endif

// Per-lane async copy of 16 bytes global -> LDS (ASYNCcnt-tracked on CDNA5).
__device__ __forceinline__ void copy_b128(const float* g, float* l) {
#ifdef ASYNC_LDS
    __builtin_amdgcn_global_load_async_to_lds_b128(
        (__attribute__((address_space(1))) gv4i*)(uintptr_t)g,
        (__attribute__((address_space(3))) gv4i*)(uintptr_t)l,
        /*offset=*/0, /*cpol=*/0);
#else
    *(v4f*)l = *(const v4f*)g;
#endif
}

__device__ __forceinline__ void wait_async_copies() {
#ifdef ASYNC_LDS
#if __has_builtin(__builtin_amdgcn_s_wait_asynccnt)
    __builtin_amdgcn_s_wait_asynccnt(0);
#else
    asm volatile("s_wait_asynccnt 0x0" ::: "memory");
#endif
#endif
}

// Stage one KB-wide K-slice of A (from X) and B into LDS buffers.
// 128 rows x 32 dwords per matrix = 1024 float4; 256 threads x 4 each.
__device__ __forceinline__ void stage_slice(const float* __restrict__ gA,
                                            const float* __restrict__ gB,
                                            float* ldsA, float* ldsB,
                                            int tid, int H)
{
#pragma unroll
    for (int i = 0; i < 4; ++i) {
        const int f   = tid + i * 256;
        const int row = f >> 3;           // 8 float4 per row
        const int c4  = (f & 7) << 2;     // dword column
        copy_b128(gA + (size_t)row * H + c4, ldsA + row * LDSROW + c4);
        copy_b128(gB + (size_t)row * H + c4, ldsB + row * LDSROW + c4);
    }
}

// ---------------------------------------------------------------------------
// Phase 1: U = X @ B^T   (fp32 WMMA 16x16x4), double-buffered async staging.
// X: [T,H] row-major, B: [H,H] row-major, U: [T,H]
// 8 waves per block in a 2(M) x 4(N) grid; each wave computes a 64x32 patch
// as 4x2 tiles of 16x16 (8 v8f accumulators).
// ---------------------------------------------------------------------------
__global__ __launch_bounds__(256)
void gemm_xBt_wmma(const float* __restrict__ X, const float* __restrict__ B,
                   float* __restrict__ U, int T, int H)
{
    __shared__ float As[2][TM * LDSROW];
    __shared__ float Bs[2][TN * LDSROW];

    const int tid   = threadIdx.x;
    const int wave  = tid >> 5;
    const int lane  = tid & 31;
    const int l16   = lane & 15;
    const int khalf = lane >> 4;          // 0: K=k,k+1   1: K=k+2,k+3

    const int tileM = blockIdx.y * TM;
    const int tileN = blockIdx.x * TN;
    const int waveM = (wave & 1) * 64;    // 2 waves across M
    const int waveN = (wave >> 1) * 32;   // 4 waves across N

    const float* gA = X + (size_t)tileM * H;   // A tile base (advance by k0)
    const float* gB = B + (size_t)tileN * H;   // B tile base

    v8f acc[4][2];
    const v8f vzero = {0.f, 0.f, 0.f, 0.f, 0.f, 0.f, 0.f, 0.f};
#pragma unroll
    for (int i = 0; i < 4; ++i)
#pragma unroll
        for (int j = 0; j < 2; ++j) acc[i][j] = vzero;

    // Prologue: stage first slice into buffer 0.
    stage_slice(gA, gB, As[0], Bs[0], tid, H);
    wait_async_copies();
    __syncthreads();

    int cur = 0;
    for (int k0 = 0; k0 < H; k0 += KB) {
        // Issue staging of the next slice into the other buffer (no wait).
        if (k0 + KB < H)
            stage_slice(gA + k0 + KB, gB + k0 + KB,
                        As[cur ^ 1], Bs[cur ^ 1], tid, H);

        // Compute on the current buffer.
        const float* as = As[cur];
        const float* bs = Bs[cur];
#pragma unroll
        for (int kk = 0; kk < KB; kk += 4) {
            const int kidx = kk + 2 * khalf;
            v2f a[4], b[2];
#pragma unroll
            for (int i = 0; i < 4; ++i)
                a[i] = *(const v2f*)(as + (waveM + i * 16 + l16) * LDSROW + kidx);
#pragma unroll
            for (int j = 0; j < 2; ++j)
                b[j] = *(const v2f*)(bs + (waveN + j * 16 + l16) * LDSROW + kidx);
#pragma unroll
            for (int i = 0; i < 4; ++i)
#pragma unroll
                for (int j = 0; j < 2; ++j)
                    acc[i][j] = __builtin_amdgcn_wmma_f32_16x16x4_f32(
                        false, a[i], false, b[j], (short)0, acc[i][j],
                        false, false);
        }

        // Drain this iteration's async copies, then one barrier per K-block.
        wait_async_copies();
        __syncthreads();
        cur ^= 1;
    }

    // C/D layout: VGPR r holds (M = r + 8*khalf, N = l16) within each 16x16 tile.
    const int rowBase = tileM + waveM + khalf * 8;
    const int colBase = tileN + waveN + l16;
#pragma unroll
    for (int i = 0; i < 4; ++i)
#pragma unroll
        for (int j = 0; j < 2; ++j) {
#pragma unroll
            for (int r = 0; r < 8; ++r) {
                const int row = rowBase + i * 16 + r;
                const int col = colBase + j * 16;
                U[(size_t)row * H + col] = acc[i][j][r];
            }
        }
}

// ---------------------------------------------------------------------------
// Phase 2a: per-(chunk,channel) local scan, in place. carry[c][ch] = h at chunk end.
// ---------------------------------------------------------------------------
__global__ __launch_bounds__(256)
void scan_local(float* __restrict__ U, const float* __restrict__ lamda,
                float* __restrict__ carry, int H)
{
    const int ch = blockIdx.x * 256 + threadIdx.x;   // channel
    const int c  = blockIdx.y;                       // chunk
    const float lam = lamda[ch];
    const size_t base = (size_t)c * CHUNK * H + ch;
    float h = 0.f;
    for (int i = 0; i < CHUNK; ++i) {
        h = fmaf(lam, h, U[base + (size_t)i * H]);
        U[base + (size_t)i * H] = h;
    }
    carry[c * H + ch] = h;
}

// ---------------------------------------------------------------------------
// Phase 2b: per-channel scan over chunk carries; hin[c][ch] = h entering chunk c.
// Chunk decay factor is lam^CHUNK via 7 squarings.
// ---------------------------------------------------------------------------
__global__ __launch_bounds__(256)
void scan_chunks(const float* __restrict__ lamda, const float* __restrict__ carry,
                 float* __restrict__ hin, int H, int nchunk)
{
    const int ch = blockIdx.x * 256 + threadIdx.x;
    const float lam = lamda[ch];
    float a = lam;
#pragma unroll
    for (int s = 0; s < 7; ++s) a *= a;              // lam^128
    float h = 0.f;
    for (int c = 0; c < nchunk; ++c) {
        hin[c * H + ch] = h;
        h = fmaf(a, h, carry[c * H + ch]);
    }
}

// ---------------------------------------------------------------------------
// Phase 2c: fixup: h_t += lam^{i+1} * h_in for t = c*CHUNK + i (chunks >= 1).
// ---------------------------------------------------------------------------
__global__ __launch_bounds__(256)
void scan_fixup(float* __restrict__ U, const float* __restrict__ lamda,
                const float* __restrict__ hin, int H)
{
    const int ch = blockIdx.x * 256 + threadIdx.x;
    const int c  = blockIdx.y + 1;                   // chunk 0 needs no fixup
    const float lam = lamda[ch];
    const float hv  = hin[c * H + ch];
    const size_t base = (size_t)c * CHUNK * H + ch;
    float p = lam;
    for (int i = 0; i < CHUNK; ++i) {
        U[base + (size_t)i * H] = fmaf(p, hv, U[base + (size_t)i * H]);
        p *= lam;
    }
}

extern "C" void kernel_launch(void* const* d_in, const int* in_sizes, int n_in,
                              void* d_out, int out_size, void* d_ws, size_t ws_size,
                              hipStream_t stream) {
    const float* X   = (const float*)d_in[0];   // x_sequence [T,H]
    const float* lam = (const float*)d_in[1];   // lamda [H]
    const float* B   = (const float*)d_in[2];   // B [H,H]
    float* out = (float*)d_out;                 // h [T,H] (holds u in-flight)

    const int H = in_sizes[1];                  // 2048
    const int T = in_sizes[0] / H;              // 8192
    const int nchunk = T / CHUNK;               // 64

    float* carry = (float*)d_ws;                // [nchunk][H]
    float* hin   = carry + (size_t)nchunk * H;  // [nchunk][H]

    dim3 gGemm(H / TN, T / TM);                 // (16, 64)
    gemm_xBt_wmma<<<gGemm, 256, 0, stream>>>(X, B, out, T, H);

    dim3 gS1(H / 256, nchunk);
    scan_local<<<gS1, 256, 0, stream>>>(out, lam, carry, H);

    scan_chunks<<<H / 256, 256, 0, stream>>>(lam, carry, hin, H, nchunk);

    dim3 gS3(H / 256, nchunk - 1);
    scan_fixup<<<gS3, 256, 0, stream>>>(out, lam, hin, H);
}